// DetBenchPredict_37314675868041
// MI455X (gfx1250) — compile-verified
//
#include <hip/hip_runtime.h>
#include <stdint.h>

// ---------------------------------------------------------------------------
// EfficientDet post-processing for MI455X (gfx1250, wave32).
// Pipeline: histogram top-k (2-level radix threshold) -> compaction ->
// box decode -> greedy NMS in LDS (staged via CDNA5 async-to-LDS copies).
// ---------------------------------------------------------------------------

#define NUM_CLASSES 90
#define TOPK        5000
#define NDET        100
#define BATCH       8
#define NBINS       4096

// ----- order-preserving float -> uint key (ascending) -----------------------
__device__ __forceinline__ unsigned okey(float f) {
    unsigned u = __float_as_uint(f);
    return (u & 0x80000000u) ? ~u : (u | 0x80000000u);
}

// ---------------------------------------------------------------------------
// Pass 1: coarse histogram over key[31:20], per image (blockIdx.y = batch).
// ---------------------------------------------------------------------------
__global__ void k_hist1(const float* __restrict__ src, unsigned* __restrict__ hist, int n4) {
    const int b = blockIdx.y;
    __shared__ unsigned h[NBINS];
    for (int i = threadIdx.x; i < NBINS; i += blockDim.x) h[i] = 0u;
    __syncthreads();
    const float4* p = (const float4*)src + (size_t)b * n4;
    const int stride = gridDim.x * blockDim.x;
    for (int i = blockIdx.x * blockDim.x + threadIdx.x; i < n4; i += stride) {
        float4 v = p[i];
        atomicAdd(&h[okey(v.x) >> 20], 1u);
        atomicAdd(&h[okey(v.y) >> 20], 1u);
        atomicAdd(&h[okey(v.z) >> 20], 1u);
        atomicAdd(&h[okey(v.w) >> 20], 1u);
    }
    __syncthreads();
    unsigned* gh = hist + (size_t)b * NBINS;
    for (int i = threadIdx.x; i < NBINS; i += blockDim.x) {
        unsigned c = h[i];
        if (c) atomicAdd(&gh[i], c);
    }
}

// ---------------------------------------------------------------------------
// Scan 1: suffix-scan from top bin; find T1 with count(>T1) < K <= count(>=T1).
// tinfo[b*4+0]=T1, tinfo[b*4+1]=count strictly above T1.
// ---------------------------------------------------------------------------
__global__ void k_scan1(const unsigned* __restrict__ hist, int* __restrict__ tinfo) {
    const int b = blockIdx.x;
    if (threadIdx.x != 0) return;
    const unsigned* h = hist + (size_t)b * NBINS;
    unsigned run = 0; int T = 0; unsigned above = 0;
    for (int bin = NBINS - 1; bin >= 0; --bin) {
        unsigned c = h[bin];
        if (run + c >= (unsigned)TOPK) { T = bin; above = run; break; }
        run += c;
    }
    tinfo[b * 4 + 0] = T;
    tinfo[b * 4 + 1] = (int)above;
}

// ---------------------------------------------------------------------------
// Pass 2: refine inside boundary bin T1 using key[19:8]. Very few atomics.
// ---------------------------------------------------------------------------
__global__ void k_hist2(const float* __restrict__ src, unsigned* __restrict__ hist2,
                        const int* __restrict__ tinfo, int n4) {
    const int b = blockIdx.y;
    const unsigned T1 = (unsigned)tinfo[b * 4 + 0];
    const float4* p = (const float4*)src + (size_t)b * n4;
    unsigned* gh = hist2 + (size_t)b * NBINS;
    const int stride = gridDim.x * blockDim.x;
    for (int i = blockIdx.x * blockDim.x + threadIdx.x; i < n4; i += stride) {
        float4 v = p[i];
        unsigned u;
        u = okey(v.x); if ((u >> 20) == T1) atomicAdd(&gh[(u >> 8) & 0xFFFu], 1u);
        u = okey(v.y); if ((u >> 20) == T1) atomicAdd(&gh[(u >> 8) & 0xFFFu], 1u);
        u = okey(v.z); if ((u >> 20) == T1) atomicAdd(&gh[(u >> 8) & 0xFFFu], 1u);
        u = okey(v.w); if ((u >> 20) == T1) atomicAdd(&gh[(u >> 8) & 0xFFFu], 1u);
    }
}

// ---------------------------------------------------------------------------
// Scan 2: find T2; init compaction counters: strict counter = 0,
// boundary counter = strictTotal (so slots never collide; cap at TOPK).
// ---------------------------------------------------------------------------
__global__ void k_scan2(const unsigned* __restrict__ hist2, int* __restrict__ tinfo,
                        unsigned* __restrict__ counters) {
    const int b = blockIdx.x;
    if (threadIdx.x != 0) return;
    const unsigned* h = hist2 + (size_t)b * NBINS;
    unsigned run = (unsigned)tinfo[b * 4 + 1];
    int T = 0; unsigned strict = run;
    for (int bin = NBINS - 1; bin >= 0; --bin) {
        unsigned c = h[bin];
        if (run + c >= (unsigned)TOPK) { T = bin; strict = run; break; }
        run += c;
    }
    tinfo[b * 4 + 2] = T;
    tinfo[b * 4 + 3] = (int)strict;
    counters[b * 2 + 0] = 0u;       // strict-accept counter
    counters[b * 2 + 1] = strict;   // boundary counter starts after strict block
}

// ---------------------------------------------------------------------------
// Pass 3: compaction. Accept key strictly above (T1,T2); boundary (==,==)
// fills remaining slots, capped at TOPK. Index math reproduces the reference
// flatten order: flat = (anchorOff + hw*9 + a) * 90 + c, where channel = a*90+c.
// ---------------------------------------------------------------------------
__global__ void k_compact(const float* __restrict__ src, const int* __restrict__ tinfo,
                          unsigned* __restrict__ counters,
                          float* __restrict__ candV, int* __restrict__ candI,
                          int n4, int HW, int anchorOff) {
    const int b = blockIdx.y;
    const unsigned T1 = (unsigned)tinfo[b * 4 + 0];
    const unsigned T2 = (unsigned)tinfo[b * 4 + 2];
    const float4* p = (const float4*)src + (size_t)b * n4;
    const int stride = gridDim.x * blockDim.x;
    for (int i = blockIdx.x * blockDim.x + threadIdx.x; i < n4; i += stride) {
        float4 v = p[i];
        float vs[4] = {v.x, v.y, v.z, v.w};
        int e0  = i * 4;
        int ch  = e0 / HW;            // HW % 4 == 0 -> all 4 share the channel
        int hw0 = e0 - ch * HW;
        int a = ch / NUM_CLASSES;
        int c = ch - a * NUM_CLASSES;
#pragma unroll
        for (int k = 0; k < 4; ++k) {
            unsigned u  = okey(vs[k]);
            unsigned b1 = u >> 20;
            if (b1 < T1) continue;
            unsigned b2 = (u >> 8) & 0xFFFu;
            if (b1 == T1 && b2 < T2) continue;
            bool strict = (b1 > T1) || (b2 > T2);
            unsigned pos = atomicAdd(&counters[b * 2 + (strict ? 0 : 1)], 1u);
            if (!strict && pos >= (unsigned)TOPK) continue;   // drop tie overflow
            candV[(size_t)b * TOPK + pos] = vs[k];
            candI[(size_t)b * TOPK + pos] = (anchorOff + (hw0 + k) * 9 + a) * NUM_CLASSES + c;
        }
    }
}

// ---------------------------------------------------------------------------
// Decode: gather box regression + anchor, decode, clip, sigmoid, class offset.
// ---------------------------------------------------------------------------
struct Ptr5 { const float* p[5]; };

__global__ void k_decode(Ptr5 boxp,
                         const float* __restrict__ anchors,
                         const float* __restrict__ img_scales,
                         const float* __restrict__ img_size,
                         const float* __restrict__ candV,
                         const int*   __restrict__ candI,
                         float* __restrict__ sig,
                         float4* __restrict__ obox,
                         float4* __restrict__ rbox,
                         int* __restrict__ cls) {
    int t = blockIdx.x * blockDim.x + threadIdx.x;
    if (t >= BATCH * TOPK) return;
    int b = t / TOPK;
    int idx = candI[t];
    float logit = candV[t];
    int g = idx / NUM_CLASSES;
    int c = idx - g * NUM_CLASSES;
    int l, off, HW;
    if      (g <  82944) { l = 0; off = 0;      HW = 9216; }
    else if (g < 103680) { l = 1; off = 82944;  HW = 2304; }
    else if (g < 108864) { l = 2; off = 103680; HW = 576;  }
    else if (g < 110160) { l = 3; off = 108864; HW = 144;  }
    else                 { l = 4; off = 110160; HW = 36;   }
    int local = g - off;
    int hw = local / 9;
    int a  = local - hw * 9;
    const float* bp = boxp.p[l] + (size_t)b * 36 * HW + hw;   // (B,36,H,W), ch=a*4+k
    float ty  = bp[(a * 4 + 0) * HW];
    float tx  = bp[(a * 4 + 1) * HW];
    float th_ = bp[(a * 4 + 2) * HW];
    float tw  = bp[(a * 4 + 3) * HW];
    float ay1 = anchors[g * 4 + 0], ax1 = anchors[g * 4 + 1];
    float ay2 = anchors[g * 4 + 2], ax2 = anchors[g * 4 + 3];
    float yca = 0.5f * (ay1 + ay2), xca = 0.5f * (ax1 + ax2);
    float ha = ay2 - ay1, wa = ax2 - ax1;
    float w  = __expf(tw)  * wa;
    float h  = __expf(th_) * ha;
    float yc = ty * ha + yca;
    float xc = tx * wa + xca;
    float x1 = xc - 0.5f * w, y1 = yc - 0.5f * h;
    float x2 = xc + 0.5f * w, y2 = yc + 0.5f * h;
    float scale = img_scales[b];
    float l0 = img_size[b * 2 + 0] / scale;
    float l1 = img_size[b * 2 + 1] / scale;
    x1 = fminf(fmaxf(x1, 0.f), l0); y1 = fminf(fmaxf(y1, 0.f), l1);
    x2 = fminf(fmaxf(x2, 0.f), l0); y2 = fminf(fmaxf(y2, 0.f), l1);
    float oc = (float)c * 10000.0f;
    rbox[t] = make_float4(x1, y1, x2, y2);
    obox[t] = make_float4(x1 + oc, y1 + oc, x2 + oc, y2 + oc);
    sig[t]  = 1.0f / (1.0f + __expf(-logit));
    cls[t]  = c;
}

// ---------------------------------------------------------------------------
// NMS: one workgroup per image. Candidates staged to LDS via CDNA5
// GLOBAL_LOAD_ASYNC_TO_LDS (tracked with ASYNCcnt), then 100 sequential
// argmax + suppress iterations entirely out of LDS. wave32 shuffles.
// ---------------------------------------------------------------------------
__global__ __launch_bounds__(1024) void k_nms(const float4* __restrict__ obox,
                                              const float4* __restrict__ rbox,
                                              const float*  __restrict__ sig,
                                              const int*    __restrict__ cls,
                                              const float*  __restrict__ img_scales,
                                              float* __restrict__ out) {
    const int b   = blockIdx.x;
    const int tid = threadIdx.x;
    __shared__ float4 sb[TOPK];                 // class-offset boxes (80 KB)
    __shared__ float  ss[TOPK];                 // mutable scores (20 KB)
    __shared__ unsigned long long red[32];
    __shared__ unsigned long long bestS;

    const float4* gob = obox + (size_t)b * TOPK;
    const float*  gs  = sig  + (size_t)b * TOPK;

    // ---- CDNA5 async copy global -> LDS; low 32 bits of a generic LDS
    // pointer are the LDS byte offset (flat aperture layout) ----
    for (int e = tid; e < TOPK; e += blockDim.x) {
        unsigned           laddr = (unsigned)(uintptr_t)&sb[e];
        unsigned long long gaddr = (unsigned long long)(uintptr_t)(gob + e);
        asm volatile("global_load_async_to_lds_b128 %0, %1, off"
                     :: "v"(laddr), "v"(gaddr) : "memory");
    }
    for (int e = tid; e < TOPK; e += blockDim.x) {
        unsigned           laddr = (unsigned)(uintptr_t)&ss[e];
        unsigned long long gaddr = (unsigned long long)(uintptr_t)(gs + e);
        asm volatile("global_load_async_to_lds_b32 %0, %1, off"
                     :: "v"(laddr), "v"(gaddr) : "memory");
    }
    asm volatile("s_wait_asynccnt 0" ::: "memory");
    __syncthreads();

    const float scale = img_scales[b];
    const int lane = tid & 31;
    const int wid  = tid >> 5;

    for (int t = 0; t < NDET; ++t) {
        // ---- argmax over ss with lowest-index tie-break ----
        unsigned long long best = 0ull;
        for (int e = tid; e < TOPK; e += blockDim.x) {
            unsigned long long k = ((unsigned long long)okey(ss[e]) << 32)
                                 | (unsigned long long)(0xFFFFFFFFu - (unsigned)e);
            if (k > best) best = k;
        }
#pragma unroll
        for (int o = 16; o > 0; o >>= 1) {
            unsigned long long other = __shfl_down(best, o, 32);
            if (other > best) best = other;
        }
        if (lane == 0) red[wid] = best;
        __syncthreads();
        if (wid == 0) {
            best = red[lane];
#pragma unroll
            for (int o = 16; o > 0; o >>= 1) {
                unsigned long long other = __shfl_down(best, o, 32);
                if (other > best) best = other;
            }
            if (lane == 0) bestS = best;
        }
        __syncthreads();
        best = bestS;
        const int i = (int)(0xFFFFFFFFu - (unsigned)(best & 0xFFFFFFFFull));

        if (tid == 0) {
            float si = ss[i];
            float* o = out + ((size_t)b * NDET + t) * 6;
            if (si > 0.0f) {
                float4 rb = rbox[(size_t)b * TOPK + i];
                int    ci = cls[(size_t)b * TOPK + i];
                float X  = rb.x * scale, Y  = rb.y * scale;
                float X2 = rb.z * scale, Y2 = rb.w * scale;
                o[0] = X; o[1] = Y; o[2] = X2 - X; o[3] = Y2 - Y;
                o[4] = si; o[5] = (float)(ci + 1);
            } else {
                o[0] = 0.f; o[1] = 0.f; o[2] = 0.f; o[3] = 0.f; o[4] = 0.f; o[5] = 0.f;
            }
        }
        const float4 bi = sb[i];
        const float areai = (bi.z - bi.x) * (bi.w - bi.y);
        __syncthreads();   // tid0's read of ss[i] before suppression writes

        for (int e = tid; e < TOPK; e += blockDim.x) {
            if (e == i) { ss[e] = -1.0f; continue; }
            float4 bj = sb[e];
            float xx1 = fmaxf(bi.x, bj.x);
            float yy1 = fmaxf(bi.y, bj.y);
            float xx2 = fminf(bi.z, bj.z);
            float yy2 = fminf(bi.w, bj.w);
            float inter = fmaxf(xx2 - xx1, 0.f) * fmaxf(yy2 - yy1, 0.f);
            float areaj = (bj.z - bj.x) * (bj.w - bj.y);
            float iou = inter / (areai + areaj - inter + 1e-8f);
            if (iou > 0.5f) ss[e] = -1.0f;
        }
        __syncthreads();
    }
}

// ---------------------------------------------------------------------------
// Host side
// ---------------------------------------------------------------------------
extern "C" void kernel_launch(void* const* d_in, const int* in_sizes, int n_in,
                              void* d_out, int out_size, void* d_ws, size_t ws_size,
                              hipStream_t stream) {
    (void)in_sizes; (void)n_in; (void)out_size; (void)ws_size;
    // setup_inputs order is interleaved: cls0,box0,cls1,box1,...,anchors,scales,size
    const float* clsL[5] = {(const float*)d_in[0], (const float*)d_in[2],
                            (const float*)d_in[4], (const float*)d_in[6],
                            (const float*)d_in[8]};
    const float* boxL[5] = {(const float*)d_in[1], (const float*)d_in[3],
                            (const float*)d_in[5], (const float*)d_in[7],
                            (const float*)d_in[9]};
    const float* anchors    = (const float*)d_in[10];
    const float* img_scales = (const float*)d_in[11];
    const float* img_size   = (const float*)d_in[12];
    float* out = (float*)d_out;

    // workspace layout (bytes)
    char* ws = (char*)d_ws;
    unsigned* hist1    = (unsigned*)(ws + 0);        // 8*4096*4 = 131072
    unsigned* hist2    = (unsigned*)(ws + 131072);   // 131072
    unsigned* counters = (unsigned*)(ws + 262144);   // 8*2*4 = 64
    int*      tinfo    = (int*)     (ws + 262208);   // 8*4*4 = 128
    float*    candV    = (float*)   (ws + 262336);   // 8*5000*4 = 160000
    int*      candI    = (int*)     (ws + 422336);   // 160000
    float*    sigp     = (float*)   (ws + 582336);   // 160000
    float4*   oboxp    = (float4*)  (ws + 742336);   // 8*5000*16 = 640000 (16B aligned)
    float4*   rboxp    = (float4*)  (ws + 1382336);  // 640000
    int*      clsI     = (int*)     (ws + 2022336);  // 160000 -> total 2182336 B

    hipMemsetAsync(d_ws, 0, 262208, stream);         // zero hist1/hist2/counters

    static const int n4s [5] = {1866240, 466560, 116640, 29160, 7290}; // 810*HW/4
    static const int HWs [5] = {9216, 2304, 576, 144, 36};
    static const int aoff[5] = {0, 82944, 103680, 108864, 110160};

    for (int l = 0; l < 5; ++l) {
        int bx = (n4s[l] + 4095) / 4096;             // ~16 float4 per thread
        k_hist1<<<dim3(bx, BATCH), 256, 0, stream>>>(clsL[l], hist1, n4s[l]);
    }
    k_scan1<<<BATCH, 32, 0, stream>>>(hist1, tinfo);
    for (int l = 0; l < 5; ++l) {
        int bx = (n4s[l] + 4095) / 4096;
        k_hist2<<<dim3(bx, BATCH), 256, 0, stream>>>(clsL[l], hist2, tinfo, n4s[l]);
    }
    k_scan2<<<BATCH, 32, 0, stream>>>(hist2, tinfo, counters);
    for (int l = 0; l < 5; ++l) {
        int bx = (n4s[l] + 4095) / 4096;
        k_compact<<<dim3(bx, BATCH), 256, 0, stream>>>(clsL[l], tinfo, counters,
                                                       candV, candI, n4s[l], HWs[l], aoff[l]);
    }
    Ptr5 bp;
    for (int l = 0; l < 5; ++l) bp.p[l] = boxL[l];
    k_decode<<<(BATCH * TOPK + 255) / 256, 256, 0, stream>>>(
        bp, anchors, img_scales, img_size, candV, candI, sigp, oboxp, rboxp, clsI);
    k_nms<<<BATCH, 1024, 0, stream>>>(oboxp, rboxp, sigp, clsI, img_scales, out);
}